// LinearEncoder_49598282334723
// MI455X (gfx1250) — compile-verified
//
#include <hip/hip_runtime.h>

typedef __attribute__((ext_vector_type(2))) float v2f;
typedef __attribute__((ext_vector_type(8))) float v8f;

#define WPB 4  // waves per block; each wave owns a 32x32 output tile

// ---------------------------------------------------------------------------
// C[N,M] = concat(A0,A1,A2)[N,Ktot] @ W[Ktot,M] (+bias) (+ReLU)
// fp32 WMMA 16x16x4. Each wave32 computes a 32x32 tile = 2x2 fragments.
// Segments are iterated at the OUTER level so all inner control flow is
// wave-uniform and A-pair loads merge into aligned b64 loads.
// ---------------------------------------------------------------------------
__global__ void gemm_wmma_f32(
    const float* __restrict__ A0, int K0,
    const float* __restrict__ A1, int K1,
    const float* __restrict__ A2, int K2,
    const float* __restrict__ W,     // [Ktot, M] row-major
    const float* __restrict__ bias,  // may be null
    float* __restrict__ C,           // [N, M]
    int N, int M, int do_relu)
{
  const int lane = threadIdx.x & 31;
  const int wave = threadIdx.x >> 5;
  const int half = lane >> 4;   // 0: lanes 0-15, 1: lanes 16-31
  const int lrow = lane & 15;

  const int row0 = blockIdx.x * 32;
  const int col0 = (blockIdx.y * WPB + wave) * 32;
  if (row0 >= N || col0 >= M) return;   // wave-uniform

  // A rows this lane feeds (two 16-row fragments), clamped for safety
  const int arowA = min(row0 +      lrow, N - 1);
  const int arowB = min(row0 + 16 + lrow, N - 1);
  // B cols this lane feeds (two 16-col fragments)
  const int bcol0 = col0 + lrow;
  const int bcol1 = col0 + 16 + lrow;

  v8f acc00 = {}, acc01 = {}, acc10 = {}, acc11 = {};

  const float* As[3] = { A0, A1, A2 };
  const int    Ks[3] = { K0, K1, K2 };
  int koff = 0;

  for (int s = 0; s < 3; ++s) {
    const int Kseg = Ks[s];
    if (Kseg == 0) break;
    const float* __restrict__ A  = As[s];
    const float* __restrict__ Ws = W + (size_t)koff * M;

    const float* __restrict__ ra = A + (size_t)arowA * Kseg;
    const float* __restrict__ rb = A + (size_t)arowB * Kseg;

#pragma unroll 4
    for (int kk = 0; kk < Kseg; kk += 4) {
      const int kb = kk + half * 2;   // even -> 8B-aligned v2f loads
      v2f a0 = *(const v2f*)(ra + kb);
      v2f a1 = *(const v2f*)(rb + kb);
      v2f b0, b1;
      b0.x = Ws[(size_t)kb       * M + bcol0];
      b0.y = Ws[(size_t)(kb + 1) * M + bcol0];
      b1.x = Ws[(size_t)kb       * M + bcol1];
      b1.y = Ws[(size_t)(kb + 1) * M + bcol1];
      acc00 = __builtin_amdgcn_wmma_f32_16x16x4_f32(false, a0, false, b0, (short)0, acc00, false, false);
      acc01 = __builtin_amdgcn_wmma_f32_16x16x4_f32(false, a0, false, b1, (short)0, acc01, false, false);
      acc10 = __builtin_amdgcn_wmma_f32_16x16x4_f32(false, a1, false, b0, (short)0, acc10, false, false);
      acc11 = __builtin_amdgcn_wmma_f32_16x16x4_f32(false, a1, false, b1, (short)0, acc11, false, false);
    }
    koff += Kseg;
  }

  const float bv0 = bias ? bias[bcol0] : 0.0f;
  const float bv1 = bias ? bias[bcol1] : 0.0f;

#pragma unroll
  for (int r = 0; r < 8; ++r) {
    const int orowA = row0 +      r + half * 8;
    const int orowB = row0 + 16 + r + half * 8;
    if (orowA < N) {
      float v0 = acc00[r] + bv0;
      float v1 = acc01[r] + bv1;
      if (do_relu) { v0 = fmaxf(v0, 0.0f); v1 = fmaxf(v1, 0.0f); }
      C[(size_t)orowA * M + bcol0] = v0;
      C[(size_t)orowA * M + bcol1] = v1;
    }
    if (orowB < N) {
      float v0 = acc10[r] + bv0;
      float v1 = acc11[r] + bv1;
      if (do_relu) { v0 = fmaxf(v0, 0.0f); v1 = fmaxf(v1, 0.0f); }
      C[(size_t)orowB * M + bcol0] = v0;
      C[(size_t)orowB * M + bcol1] = v1;
    }
  }
}

// ---------------------------------------------------------------------------
// Degree / normalization kernels
// ---------------------------------------------------------------------------
__global__ void deg_init(float* __restrict__ deg, int N) {
  int i = blockIdx.x * blockDim.x + threadIdx.x;
  if (i < N) deg[i] = 1.0f;  // self-loop contributes 1 to every node
}

__global__ void deg_count(const long long* __restrict__ dst,
                          float* __restrict__ deg, int E) {
  int e = blockIdx.x * blockDim.x + threadIdx.x;
  if (e < E) atomicAdd(&deg[(int)dst[e]], 1.0f);
}

__global__ void deg_rsqrt(const float* __restrict__ deg,
                          float* __restrict__ dinv, int N) {
  int i = blockIdx.x * blockDim.x + threadIdx.x;
  if (i < N) dinv[i] = rsqrtf(deg[i]);  // deg >= 1 always (self-loop)
}

// out[n,c] = bias[c] + xw[n,c] * dinv[n]^2   (self-loop term; also initializes)
__global__ void gcn_self_init(const float* __restrict__ xw,
                              const float* __restrict__ dinv,
                              const float* __restrict__ bias,
                              float* __restrict__ out, int N, int Cw) {
  int i = blockIdx.x * blockDim.x + threadIdx.x;
  if (i >= N * Cw) return;
  int n = i / Cw;
  int c = i - n * Cw;
  float di = dinv[n];
  out[i] = bias[c] + xw[i] * (di * di);
}

// out[dst] += xw[src] * dinv[src]*dinv[dst]   (float4 gather, f32 atomics)
__global__ void gcn_scatter(const long long* __restrict__ src,
                            const long long* __restrict__ dst,
                            const float* __restrict__ xw,
                            const float* __restrict__ dinv,
                            float* __restrict__ out, int E, int Cw) {
  int e = blockIdx.x * blockDim.y + threadIdx.y;
  if (e >= E) return;
  int s = (int)src[e];
  int d = (int)dst[e];
  float nrm = dinv[s] * dinv[d];
  int c = threadIdx.x * 4;
  const float4 v = *(const float4*)(xw + (size_t)s * Cw + c);
  float* o = out + (size_t)d * Cw + c;
  atomicAdd(o + 0, v.x * nrm);
  atomicAdd(o + 1, v.y * nrm);
  atomicAdd(o + 2, v.z * nrm);
  atomicAdd(o + 3, v.w * nrm);
}

// ---------------------------------------------------------------------------
extern "C" void kernel_launch(void* const* d_in, const int* in_sizes, int n_in,
                              void* d_out, int out_size, void* d_ws, size_t ws_size,
                              hipStream_t stream) {
  const int D = 128, Cc = 128;
  const int N = in_sizes[0] / D;
  const int E = in_sizes[2] / 2;

  const float* x_self = (const float*)d_in[0];
  const float* x_nb   = (const float*)d_in[1];
  const long long* ei = (const long long*)d_in[2];
  const long long* src = ei;
  const long long* dst = ei + E;
  const float* W_in_self  = (const float*)d_in[3];
  const float* b_in_self  = (const float*)d_in[4];
  const float* W_out_self = (const float*)d_in[5];
  const float* b_out_self = (const float*)d_in[6];
  const float* Wg1 = (const float*)d_in[7];
  const float* bg1 = (const float*)d_in[8];
  const float* Wg2 = (const float*)d_in[9];
  const float* bg2 = (const float*)d_in[10];
  const float* W_out = (const float*)d_in[11];
  const float* b_out = (const float*)d_in[12];

  // workspace layout (floats)
  float* ws  = (float*)d_ws;
  float* xw1 = ws;                         // [N, 2C]
  float* g1  = xw1 + (size_t)N * 2 * Cc;   // [N, 2C]
  float* xw2 = g1  + (size_t)N * 2 * Cc;   // [N, C]
  float* g2  = xw2 + (size_t)N * Cc;       // [N, C]
  float* deg = g2  + (size_t)N * Cc;       // [N]
  float* dinv = deg + N;                   // [N]
  float* t1  = xw1;                        // alias: reused after scatter1

  float* out_l1 = (float*)d_out;               // [N, C]
  float* out_x2 = out_l1 + (size_t)N * Cc;     // [N, C]

  auto gemm = [&](const float* A0, int K0, const float* A1, int K1,
                  const float* A2, int K2, const float* W, const float* bias,
                  float* Cd, int M, int relu) {
    dim3 grid((N + 31) / 32, (M / 32 + WPB - 1) / WPB);
    dim3 block(32 * WPB);
    hipLaunchKernelGGL(gemm_wmma_f32, grid, block, 0, stream,
                       A0, K0, A1, K1, A2, K2, W, bias, Cd, N, M, relu);
  };

  // --- degree normalization ---
  hipLaunchKernelGGL(deg_init,  dim3((N + 255) / 256), dim3(256), 0, stream, deg, N);
  hipLaunchKernelGGL(deg_count, dim3((E + 255) / 256), dim3(256), 0, stream, dst, deg, E);
  hipLaunchKernelGGL(deg_rsqrt, dim3((N + 255) / 256), dim3(256), 0, stream, deg, dinv, N);

  // --- GCN layer 1: g1 = scatter(x_nb @ Wg1) + bg1 ---
  gemm(x_nb, D, nullptr, 0, nullptr, 0, Wg1, nullptr, xw1, 2 * Cc, 0);
  hipLaunchKernelGGL(gcn_self_init, dim3((N * 2 * Cc + 255) / 256), dim3(256), 0, stream,
                     xw1, dinv, bg1, g1, N, 2 * Cc);
  {
    dim3 block(2 * Cc / 4, 256 / (2 * Cc / 4));          // 64 x 4
    dim3 grid((E + block.y - 1) / block.y);
    hipLaunchKernelGGL(gcn_scatter, grid, block, 0, stream, src, dst, xw1, dinv, g1, E, 2 * Cc);
  }

  // --- GCN layer 2: g2 = scatter(g1 @ Wg2) + bg2 ---
  gemm(g1, 2 * Cc, nullptr, 0, nullptr, 0, Wg2, nullptr, xw2, Cc, 0);
  hipLaunchKernelGGL(gcn_self_init, dim3((N * Cc + 255) / 256), dim3(256), 0, stream,
                     xw2, dinv, bg2, g2, N, Cc);
  {
    dim3 block(Cc / 4, 256 / (Cc / 4));                  // 32 x 8
    dim3 grid((E + block.y - 1) / block.y);
    hipLaunchKernelGGL(gcn_scatter, grid, block, 0, stream, src, dst, xw2, dinv, g2, E, Cc);
  }

  // --- dense self branch (t1 safely aliases xw1: scatter1 already consumed it) ---
  gemm(x_self, D, nullptr, 0, nullptr, 0, W_in_self, b_in_self, t1, 2 * Cc, 1);
  gemm(x_self, D, t1, 2 * Cc, nullptr, 0, W_out_self, b_out_self, out_l1, Cc, 0);

  // --- output head: x2 = [x_nb | g1 | g2] @ W_out + b_out ---
  gemm(x_nb, D, g1, 2 * Cc, g2, Cc, W_out, b_out, out_x2, Cc, 0);
}